// DecoderRNN_91079076479527
// MI455X (gfx1250) — compile-verified
//
#include <hip/hip_runtime.h>
#include <math.h>

#define BB 32
#define HH 1024
#define VV 32000

typedef __attribute__((ext_vector_type(2))) float v2f;
typedef __attribute__((ext_vector_type(8))) float v8f;

__device__ __forceinline__ v8f wmma4(v2f a, v2f b, v8f c) {
    // D = A(16x4) * B(4x16) + C(16x16), fp32, wave32
    return __builtin_amdgcn_wmma_f32_16x16x4_f32(false, a, false, b, (short)0, c,
                                                 false, false);
}

__device__ __forceinline__ v2f ld2(const float* p) {
    return *(const v2f*)p;
}
__device__ __forceinline__ v2f ld2_nt(const float* p) {
    return __builtin_nontemporal_load((const v2f*)p);
}

// ---------------------------------------------------------------------------
// init: h buffers from `hidden` (2,B,H); dec from `inputs`
// ---------------------------------------------------------------------------
__global__ void init_kernel(const float* __restrict__ hidden,
                            const int* __restrict__ inputs,
                            float* __restrict__ hf0, float* __restrict__ hb0,
                            int* __restrict__ dec) {
    int i = blockIdx.x * 256 + threadIdx.x;
    if (i < BB * HH)          hf0[i]           = hidden[i];
    else if (i < 2 * BB * HH) hb0[i - BB * HH] = hidden[i];
    if (i < BB) dec[i] = inputs[i];
}

// ---------------------------------------------------------------------------
// Fused embedding + GRU cell for one direction / one 16-col strip of h.
// grid = (H/16, 2), block = 192 (6 waves). Wave g computes gate g for BOTH
// M-tiles (batch rows 0..15 and 16..31): the weight (B) fragment is loaded
// ONCE per k-step and feeds two v_wmma_f32_16x16x4_f32 — register-level
// reuse of the weight stream, since B=32 makes this bandwidth-bound.
// Tiles staged to LDS, then the gate nonlinearity + blend is elementwise.
// ---------------------------------------------------------------------------
__global__ __launch_bounds__(192) void gru_step_kernel(
    const int* __restrict__ dec, const float* __restrict__ emb,
    const float* __restrict__ hcur_f, const float* __restrict__ hcur_b,
    float* __restrict__ hnew_f, float* __restrict__ hnew_b,
    const float* __restrict__ w_ih_f, const float* __restrict__ w_hh_f,
    const float* __restrict__ b_ih_f, const float* __restrict__ b_hh_f,
    const float* __restrict__ w_ih_b, const float* __restrict__ w_hh_b,
    const float* __restrict__ b_ih_b, const float* __restrict__ b_hh_b) {
    const int strip = blockIdx.x;   // 0..63
    const int dir   = blockIdx.y;   // 0 = forward, 1 = backward
    const float* hcur = dir ? hcur_b : hcur_f;
    float*       hnew = dir ? hnew_b : hnew_f;
    const float* w_ih = dir ? w_ih_b : w_ih_f;
    const float* w_hh = dir ? w_hh_b : w_hh_f;
    const float* b_ih = dir ? b_ih_b : b_ih_f;
    const float* b_hh = dir ? b_hh_b : b_hh_f;

    const int tid   = threadIdx.x;
    const int g     = tid >> 5;           // wave id == gate 0..5
    const int lane  = tid & 31;
    const int lrow  = lane & 15;
    const int koff  = (lane >> 4) << 1;   // 0 or 2
    const int jbase = strip * 16;

    __shared__ float tiles[12][16][16];   // [gate*2 + mtile], 12 KB

    // A rows for this lane: M = lrow (tile 0) and 16 + lrow (tile 1)
    const float* arow0;
    const float* arow1;
    const float* W;
    if (g < 3) {
        arow0 = emb + (size_t)dec[lrow]      * HH;
        arow1 = emb + (size_t)dec[16 + lrow] * HH;
        W = w_ih;
    } else {
        arow0 = hcur + (size_t)lrow        * HH;
        arow1 = hcur + (size_t)(16 + lrow) * HH;
        W = w_hh;
    }
    const int    wr   = (g % 3) * HH + jbase + lrow;
    const float* wrow = W + (size_t)wr * HH;

    v8f acc0 = {};
    v8f acc1 = {};
    #pragma unroll 4
    for (int k = 0; k < HH; k += 4) {
        v2f b  = ld2(wrow + k + koff);    // shared weight fragment
        v2f a0 = ld2(arow0 + k + koff);
        v2f a1 = ld2(arow1 + k + koff);
        acc0 = wmma4(a0, b, acc0);
        acc1 = wmma4(a1, b, acc1);
    }
    const int chalf = (lane >> 4) * 8;
    #pragma unroll
    for (int r = 0; r < 8; ++r) {
        tiles[g * 2 + 0][chalf + r][lrow] = acc0[r];
        tiles[g * 2 + 1][chalf + r][lrow] = acc1[r];
    }
    __syncthreads();

    // elementwise gates: 32 rows x 16 cols = 512 outputs
    for (int o = tid; o < 512; o += 192) {
        const int m = o >> 4, c = o & 15;
        const int mt = m >> 4, mr = m & 15;
        const int col = jbase + c;
        float i_r = tiles[0 * 2 + mt][mr][c] + b_ih[col];
        float i_z = tiles[1 * 2 + mt][mr][c] + b_ih[HH + col];
        float i_n = tiles[2 * 2 + mt][mr][c] + b_ih[2 * HH + col];
        float h_r = tiles[3 * 2 + mt][mr][c] + b_hh[col];
        float h_z = tiles[4 * 2 + mt][mr][c] + b_hh[HH + col];
        float h_n = tiles[5 * 2 + mt][mr][c] + b_hh[2 * HH + col];
        float r = 1.f / (1.f + expf(-(i_r + h_r)));
        float z = 1.f / (1.f + expf(-(i_z + h_z)));
        float n = tanhf(i_n + r * h_n);
        float hp = hcur[(size_t)m * HH + col];
        hnew[(size_t)m * HH + col] = (1.f - z) * n + z * hp;
    }
}

// ---------------------------------------------------------------------------
// logits = concat(h_f,h_b)(32x2048) @ w_out^T (2048x32000) + b_out.
// grid = 250, block = 256 (8 waves); each wave owns one N-strip and BOTH
// M-tiles: the w_out (B) fragment is loaded once per k-step (NT hint — this
// 262MB/step stream must not thrash L2 where the GRU weights live) and feeds
// two WMMAs. K=2048 split as 1024 from h_f then 1024 from h_b.
// ---------------------------------------------------------------------------
__global__ __launch_bounds__(256) void logits_kernel(
    const float* __restrict__ hf, const float* __restrict__ hb,
    const float* __restrict__ w_out, const float* __restrict__ b_out,
    float* __restrict__ logits) {
    const int tid   = threadIdx.x;
    const int wave  = tid >> 5;
    const int lane  = tid & 31;
    const int lrow  = lane & 15;
    const int koff  = (lane >> 4) << 1;
    const int strip = blockIdx.x * 8 + wave;         // 0..1999
    const int n     = strip * 16 + lrow;             // output column (B frag row)

    const float* af0  = hf + (size_t)lrow        * HH;
    const float* af1  = hf + (size_t)(16 + lrow) * HH;
    const float* ab0  = hb + (size_t)lrow        * HH;
    const float* ab1  = hb + (size_t)(16 + lrow) * HH;
    const float* wrow = w_out + (size_t)n * (2 * HH);

    v8f acc0 = {};
    v8f acc1 = {};
    #pragma unroll 4
    for (int k = 0; k < HH; k += 4) {
        v2f b  = ld2_nt(wrow + k + koff);
        v2f a0 = ld2(af0 + k + koff);
        v2f a1 = ld2(af1 + k + koff);
        acc0 = wmma4(a0, b, acc0);
        acc1 = wmma4(a1, b, acc1);
    }
    #pragma unroll 4
    for (int k = 0; k < HH; k += 4) {
        v2f b  = ld2_nt(wrow + HH + k + koff);
        v2f a0 = ld2(ab0 + k + koff);
        v2f a1 = ld2(ab1 + k + koff);
        acc0 = wmma4(a0, b, acc0);
        acc1 = wmma4(a1, b, acc1);
    }

    const int chalf = (lane >> 4) * 8;
    const float bias = b_out[n];
    #pragma unroll
    for (int r = 0; r < 8; ++r) {
        const int row = chalf + r;
        logits[(size_t)row * VV + n]        = acc0[r] + bias;
        logits[(size_t)(16 + row) * VV + n] = acc1[r] + bias;
    }
}

// ---------------------------------------------------------------------------
// Per batch row: max + first-occurrence argmax + logsumexp. grid=B, block=256.
// ---------------------------------------------------------------------------
__global__ __launch_bounds__(256) void reduce_kernel(
    const float* __restrict__ logits, float* __restrict__ logZ,
    int* __restrict__ dec) {
    const int b   = blockIdx.x;
    const int tid = threadIdx.x;
    const float* row = logits + (size_t)b * VV;
    __shared__ float smax[256];
    __shared__ int   sarg[256];
    __shared__ float ssum[256];

    float bm = -__FLT_MAX__;
    int   ba = 0x7fffffff;
    for (int i = tid; i < VV; i += 256) {
        float v = row[i];
        if (v > bm || (v == bm && i < ba)) { bm = v; ba = i; }
    }
    smax[tid] = bm; sarg[tid] = ba;
    __syncthreads();
    for (int s = 128; s > 0; s >>= 1) {
        if (tid < s) {
            float v2 = smax[tid + s]; int a2 = sarg[tid + s];
            if (v2 > smax[tid] || (v2 == smax[tid] && a2 < sarg[tid])) {
                smax[tid] = v2; sarg[tid] = a2;
            }
        }
        __syncthreads();
    }
    const float rmax = smax[0];

    float ls = 0.f;
    for (int i = tid; i < VV; i += 256) ls += expf(row[i] - rmax);
    ssum[tid] = ls;
    __syncthreads();
    for (int s = 128; s > 0; s >>= 1) {
        if (tid < s) ssum[tid] += ssum[tid + s];
        __syncthreads();
    }
    if (tid == 0) {
        logZ[b] = rmax + logf(ssum[0]);
        dec[b]  = sarg[0];
    }
}

// ---------------------------------------------------------------------------
// logp[b][t][v] = logits[b][v] - logZ[b]
// ---------------------------------------------------------------------------
__global__ __launch_bounds__(256) void logp_kernel(
    const float* __restrict__ logits, const float* __restrict__ logZ,
    float* __restrict__ out, int t, int T) {
    const int v = blockIdx.x * 256 + threadIdx.x;
    const int b = blockIdx.y;
    if (v < VV)
        out[((size_t)b * T + t) * VV + v] = logits[(size_t)b * VV + v] - logZ[b];
}

// ---------------------------------------------------------------------------
extern "C" void kernel_launch(void* const* d_in, const int* in_sizes, int n_in,
                              void* d_out, int out_size, void* d_ws, size_t ws_size,
                              hipStream_t stream) {
    const int*   inputs = (const int*)  d_in[0];
    const float* hidden = (const float*)d_in[1];
    const float* emb    = (const float*)d_in[2];
    const float* w_ih_f = (const float*)d_in[3];
    const float* w_hh_f = (const float*)d_in[4];
    const float* b_ih_f = (const float*)d_in[5];
    const float* b_hh_f = (const float*)d_in[6];
    const float* w_ih_b = (const float*)d_in[7];
    const float* w_hh_b = (const float*)d_in[8];
    const float* b_ih_b = (const float*)d_in[9];
    const float* b_hh_b = (const float*)d_in[10];
    const float* w_out  = (const float*)d_in[11];
    const float* b_out  = (const float*)d_in[12];
    float* out = (float*)d_out;
    const int T = out_size / (BB * VV);

    // workspace layout
    char* ws = (char*)d_ws;
    const size_t HBYTES = (size_t)BB * HH * sizeof(float);   // 128 KB
    float* hfb[2] = { (float*)(ws + 0 * HBYTES), (float*)(ws + 1 * HBYTES) };
    float* hbb[2] = { (float*)(ws + 2 * HBYTES), (float*)(ws + 3 * HBYTES) };
    float* logits = (float*)(ws + 4 * HBYTES);
    float* logZ   = (float*)(ws + 4 * HBYTES + (size_t)BB * VV * sizeof(float));
    int*   dec    = (int*)  (ws + 4 * HBYTES + (size_t)BB * VV * sizeof(float) + 256);

    init_kernel<<<(2 * BB * HH + 255) / 256, 256, 0, stream>>>(
        hidden, inputs, hfb[0], hbb[0], dec);

    for (int t = 0; t < T; ++t) {
        const float* hf_c = hfb[t & 1];
        const float* hb_c = hbb[t & 1];
        float* hf_n = hfb[(t + 1) & 1];
        float* hb_n = hbb[(t + 1) & 1];

        gru_step_kernel<<<dim3(HH / 16, 2), 192, 0, stream>>>(
            dec, emb, hf_c, hb_c, hf_n, hb_n,
            w_ih_f, w_hh_f, b_ih_f, b_hh_f,
            w_ih_b, w_hh_b, b_ih_b, b_hh_b);

        logits_kernel<<<dim3(VV / 128), 256, 0, stream>>>(
            hf_n, hb_n, w_out, b_out, logits);

        reduce_kernel<<<dim3(BB), 256, 0, stream>>>(logits, logZ, dec);

        logp_kernel<<<dim3((VV + 255) / 256, BB), 256, 0, stream>>>(
            logits, logZ, out, t, T);
    }
}